// CommunityGNN_56530359550020
// MI455X (gfx1250) — compile-verified
//
#include <hip/hip_runtime.h>
#include <hip/hip_bf16.h>

typedef __attribute__((ext_vector_type(2))) float v2f;
typedef __attribute__((ext_vector_type(8))) float v8f;

// ---------------------------------------------------------------------------
// Utility: zero a range of 32-bit words
// ---------------------------------------------------------------------------
__global__ void gnn_zero_u32(unsigned* __restrict__ p, size_t nwords) {
  size_t i = (size_t)blockIdx.x * blockDim.x + threadIdx.x;
  size_t stride = (size_t)gridDim.x * blockDim.x;
  for (; i < nwords; i += stride) p[i] = 0u;
}

// ---------------------------------------------------------------------------
// In-degree of dst (self-loop added later as +1)
// ---------------------------------------------------------------------------
__global__ void gnn_degree(const int* __restrict__ dst, int* __restrict__ deg, int E) {
  int i = blockIdx.x * blockDim.x + threadIdx.x;
  if (i < E) atomicAdd(&deg[dst[i]], 1);
}

__global__ void gnn_dinv(const int* __restrict__ deg, float* __restrict__ dinv, int Nn) {
  int i = blockIdx.x * blockDim.x + threadIdx.x;
  if (i < Nn) dinv[i] = rsqrtf((float)(deg[i] + 1));  // +1 = self loop
}

// ---------------------------------------------------------------------------
// C[M,64] = A[M,K] @ B[K,64], row-major f32, via V_WMMA_F32_16X16X4_F32.
// One wave computes a 16x64 strip: the A fragment is loaded once per K-step
// and reused by 4 independent WMMAs (distinct accumulators -> no D->A/B
// hazard stalls). B is staged in LDS once per block (pitch 65 floats to
// avoid the 2-way bank conflict between lane halves reading rows k / k+2).
// A-frag layout: lanes 0-15 -> K = {k,k+1}; lanes 16-31 -> K = {k+2,k+3}.
// C/D layout: c[i] is row (strip*16 + i + 8*(lane/16)), col (16*j + lane%16).
// ---------------------------------------------------------------------------
#define BPITCH 65
__global__ __launch_bounds__(256)
void gnn_gemm_wmma_f32_n64(const float* __restrict__ A, const float* __restrict__ B,
                           float* __restrict__ C, int M, int K) {
  __shared__ float Bs[128 * BPITCH];          // up to K=128 rows of 64 (+1 pad)
  int tid = threadIdx.x;
  for (int i = tid; i < K * 64; i += 256) {
    int r = i >> 6, c = i & 63;
    Bs[r * BPITCH + c] = B[i];
  }
  __syncthreads();

  int wave = (blockIdx.x * blockDim.x + tid) >> 5;
  int lane = tid & 31;
  int Mtiles = (M + 15) >> 4;
  if (wave >= Mtiles) return;

  int lm = lane & 15;     // M index (A) / N index within tile (B, C)
  int lh = lane >> 4;     // lane-half selects K pair
  int mrow = wave * 16 + lm;
  const float* Arow = A + (size_t)(mrow < M ? mrow : M - 1) * K;

  v8f c0 = {0.f,0.f,0.f,0.f,0.f,0.f,0.f,0.f};
  v8f c1 = c0, c2 = c0, c3 = c0;
  for (int k = 0; k < K; k += 4) {
    v2f a = *(const v2f*)(Arow + k + 2 * lh);          // contiguous pair, 8B load
    const float* Br0 = &Bs[(k + 2 * lh) * BPITCH + lm];
    const float* Br1 = Br0 + BPITCH;
    v2f b0, b1, b2, b3;
    b0.x = Br0[0];  b0.y = Br1[0];
    b1.x = Br0[16]; b1.y = Br1[16];
    b2.x = Br0[32]; b2.y = Br1[32];
    b3.x = Br0[48]; b3.y = Br1[48];
    c0 = __builtin_amdgcn_wmma_f32_16x16x4_f32(false, a, false, b0, (short)0, c0, false, false);
    c1 = __builtin_amdgcn_wmma_f32_16x16x4_f32(false, a, false, b1, (short)0, c1, false, false);
    c2 = __builtin_amdgcn_wmma_f32_16x16x4_f32(false, a, false, b2, (short)0, c2, false, false);
    c3 = __builtin_amdgcn_wmma_f32_16x16x4_f32(false, a, false, b3, (short)0, c3, false, false);
  }

  float* Crow = C + (size_t)(wave * 16) * 64 + lm;
#pragma unroll
  for (int i = 0; i < 8; ++i) {
    int r = i + 8 * lh;
    if (wave * 16 + r < M) {
      size_t o = (size_t)r * 64;
      Crow[o]      = c0[i];
      Crow[o + 16] = c1[i];
      Crow[o + 32] = c2[i];
      Crow[o + 48] = c3[i];
    }
  }
}

// ---------------------------------------------------------------------------
// Normalized scatter-add: acc[d] += t[s] * dinv[s]*dinv[d], over E edges plus
// N self-loops (edge index e >= E -> s = d = e - E). One wave per edge,
// 2 features per lane (H = 64). Hardware f32 atomics (L2-resident).
// ---------------------------------------------------------------------------
__global__ __launch_bounds__(256)
void gnn_scatter(const int* __restrict__ src, const int* __restrict__ dst,
                 const float* __restrict__ dinv, const float* __restrict__ t,
                 float* __restrict__ acc, int E, int Nn) {
  int wave = (blockIdx.x * blockDim.x + threadIdx.x) >> 5;
  int lane = threadIdx.x & 31;
  int total = E + Nn;
  if (wave >= total) return;
  int s, d;
  if (wave < E) { s = src[wave]; d = dst[wave]; }
  else          { s = wave - E; d = s; }
  const float* ts = t + (size_t)s * 64;
  __builtin_prefetch(ts + lane, 0, 3);               // global_prefetch_b8
  float nrm = dinv[s] * dinv[d];
  float* ad = acc + (size_t)d * 64;
  unsafeAtomicAdd(&ad[lane],      ts[lane]      * nrm);
  unsafeAtomicAdd(&ad[lane + 32], ts[lane + 32] * nrm);
}

// ---------------------------------------------------------------------------
// h = relu(h + b[f])   (H = 64 features)
// ---------------------------------------------------------------------------
__global__ void gnn_bias_relu(float* __restrict__ h, const float* __restrict__ b, size_t n) {
  size_t i = (size_t)blockIdx.x * blockDim.x + threadIdx.x;
  if (i < n) {
    float v = h[i] + b[i & 63];
    h[i] = v > 0.f ? v : 0.f;
  }
}

// ---------------------------------------------------------------------------
// z = (emb + b2) / max(||emb + b2||, 1e-12), in place; svec += sum_i z_i
// One wave per node (2 feats/lane), block-level LDS accumulation of svec.
// total similarity sum later = ||svec||^2  (since sum_ij z_i . z_j).
// ---------------------------------------------------------------------------
__global__ __launch_bounds__(256)
void gnn_normalize(float* __restrict__ emb, const float* __restrict__ b2,
                   float* __restrict__ svec, int Nn) {
  __shared__ float sacc[64];
  int tid = threadIdx.x;
  if (tid < 64) sacc[tid] = 0.f;
  __syncthreads();

  int wave = (blockIdx.x * blockDim.x + tid) >> 5;
  int lane = tid & 31;
  if (wave < Nn) {
    float* e = emb + (size_t)wave * 64;
    float e0 = e[lane]      + b2[lane];
    float e1 = e[lane + 32] + b2[lane + 32];
    float ss = e0 * e0 + e1 * e1;
#pragma unroll
    for (int off = 16; off > 0; off >>= 1) ss += __shfl_xor(ss, off);
    float inv = 1.0f / fmaxf(sqrtf(ss), 1e-12f);
    float z0 = e0 * inv, z1 = e1 * inv;
    e[lane] = z0;
    e[lane + 32] = z1;
    atomicAdd(&sacc[lane], z0);        // ds atomic, low contention
    atomicAdd(&sacc[lane + 32], z1);
  }
  __syncthreads();
  if (tid < 64) unsafeAtomicAdd(&svec[tid], sacc[tid]);
}

// ---------------------------------------------------------------------------
// Positive-pair similarity over DISTINCT original edges.
// Bitmask dedup replicates jnp .at[].set(1) semantics (duplicates count once).
// One wave per edge: lane0 claims the (s,d) bit with atomicOr; if first,
// 64-wide dot product wave-reduced and accumulated.
// ---------------------------------------------------------------------------
__global__ __launch_bounds__(256)
void gnn_pos(const int* __restrict__ src, const int* __restrict__ dst,
             const float* __restrict__ z, unsigned* __restrict__ mask,
             float* __restrict__ scal, int E, int Nn, int use_mask) {
  int wave = (blockIdx.x * blockDim.x + threadIdx.x) >> 5;
  int lane = threadIdx.x & 31;
  if (wave >= E) return;
  int s = src[wave], d = dst[wave];
  int first = 1;
  if (use_mask) {
    if (lane == 0) {
      int idx = s * Nn + d;                 // < 1.44e8, fits in int
      unsigned wrd = (unsigned)idx >> 5;
      unsigned bit = (unsigned)idx & 31u;
      unsigned old = atomicOr(&mask[wrd], 1u << bit);
      first = ((old >> bit) & 1u) == 0u;
    }
    first = __shfl(first, 0);
  }
  if (!first) return;
  const float* zs = z + (size_t)s * 64;
  const float* zd = z + (size_t)d * 64;
  float acc = zs[lane] * zd[lane] + zs[lane + 32] * zd[lane + 32];
#pragma unroll
  for (int off = 16; off > 0; off >>= 1) acc += __shfl_xor(acc, off);
  if (lane == 0) {
    unsafeAtomicAdd(&scal[0], acc);
    atomicAdd((int*)scal + 1, 1);
  }
}

// ---------------------------------------------------------------------------
// loss = -pos_sum/pos_cnt + (||svec||^2 - pos_sum)/(N^2 - pos_cnt)
// ---------------------------------------------------------------------------
__global__ void gnn_finalize(const float* __restrict__ svec, const float* __restrict__ scal,
                             float* __restrict__ out, int Nn) {
  __shared__ float red[64];
  int tid = threadIdx.x;
  float v = svec[tid];
  red[tid] = v * v;
  __syncthreads();
  if (tid == 0) {
    float total = 0.f;
    for (int i = 0; i < 64; ++i) total += red[i];
    float pos_sum = scal[0];
    float pos_cnt = (float)((const int*)scal)[1];
    float n2 = (float)Nn * (float)Nn;
    out[0] = -(pos_sum / pos_cnt) + (total - pos_sum) / (n2 - pos_cnt);
  }
}

// ---------------------------------------------------------------------------
// Launcher
// ---------------------------------------------------------------------------
extern "C" void kernel_launch(void* const* d_in, const int* in_sizes, int n_in,
                              void* d_out, int out_size, void* d_ws, size_t ws_size,
                              hipStream_t stream) {
  const float* x  = (const float*)d_in[0];
  const int*   ei = (const int*)d_in[1];
  const float* W1 = (const float*)d_in[2];
  const float* b1 = (const float*)d_in[3];
  const float* W2 = (const float*)d_in[4];
  const float* b2 = (const float*)d_in[5];
  float* out = (float*)d_out;

  const int F_IN = 128, Hf = 64;
  const int Nn = in_sizes[0] / F_IN;   // 12000
  const int E  = in_sizes[1] / 2;      // 384000
  const int* src = ei;
  const int* dst = ei + E;

  // Workspace layout
  char* w = (char*)d_ws;
  size_t nh = (size_t)Nn * Hf;
  float* t1   = (float*)w; w += nh * 4;
  float* acc1 = (float*)w; w += nh * 4;            // -> h1 after bias+relu
  float* t2   = (float*)w; w += nh * 4;
  float* acc2 = (float*)w; w += nh * 4;            // -> emb -> z (in place)
  float* dinv = (float*)w; w += (size_t)Nn * 4;
  int*   deg  = (int*)w;   w += (size_t)Nn * 4;
  float* svec = (float*)w; w += 64 * 4;
  float* scal = (float*)w; w += 8 * 4;             // [0]=pos_sum f32, [1]=pos_cnt i32
  unsigned* mask = (unsigned*)w;
  size_t maskWords = ((size_t)Nn * Nn + 31) / 32;
  size_t baseBytes = (size_t)(w - (char*)d_ws);
  int use_mask = (ws_size >= baseBytes + maskWords * 4) ? 1 : 0;

  const int T = 256;
  auto blocks = [](size_t n, int t) { return (unsigned)((n + (size_t)t - 1) / t); };

  // Zero accumulators / counters / bitmask (harness does not re-zero between calls)
  gnn_zero_u32<<<blocks(nh, T), T, 0, stream>>>((unsigned*)acc1, nh);
  gnn_zero_u32<<<blocks(nh, T), T, 0, stream>>>((unsigned*)acc2, nh);
  size_t tailWords = (size_t)Nn + 64 + 8;          // deg + svec + scal contiguous
  gnn_zero_u32<<<blocks(tailWords, T), T, 0, stream>>>((unsigned*)deg, tailWords);
  if (use_mask)
    gnn_zero_u32<<<blocks(maskWords, T), T, 0, stream>>>(mask, maskWords);

  // Degree (dst in-degree) and D^{-1/2}
  gnn_degree<<<blocks(E, T), T, 0, stream>>>(dst, deg, E);
  gnn_dinv<<<blocks(Nn, T), T, 0, stream>>>(deg, dinv, Nn);

  // Layer 1: t1 = x @ W1 ; acc1 = Ahat t1 ; h1 = relu(acc1 + b1)
  {
    size_t waves = (size_t)((Nn + 15) / 16);       // one wave per 16x64 strip
    gnn_gemm_wmma_f32_n64<<<blocks(waves * 32, T), T, 0, stream>>>(x, W1, t1, Nn, F_IN);
  }
  {
    size_t waves = (size_t)E + Nn;
    gnn_scatter<<<blocks(waves * 32, T), T, 0, stream>>>(src, dst, dinv, t1, acc1, E, Nn);
  }
  gnn_bias_relu<<<blocks(nh, T), T, 0, stream>>>(acc1, b1, nh);

  // Layer 2: t2 = h1 @ W2 ; acc2 = Ahat t2 ; emb = acc2 + b2 (folded into normalize)
  {
    size_t waves = (size_t)((Nn + 15) / 16);
    gnn_gemm_wmma_f32_n64<<<blocks(waves * 32, T), T, 0, stream>>>(acc1, W2, t2, Nn, Hf);
  }
  {
    size_t waves = (size_t)E + Nn;
    gnn_scatter<<<blocks(waves * 32, T), T, 0, stream>>>(src, dst, dinv, t2, acc2, E, Nn);
  }

  // Normalize rows -> z (in place), accumulate svec = sum_i z_i
  gnn_normalize<<<blocks((size_t)Nn * 32, T), T, 0, stream>>>(acc2, b2, svec, Nn);

  // Positive-edge similarity with dedup
  gnn_pos<<<blocks((size_t)E * 32, T), T, 0, stream>>>(src, dst, acc2, mask, scal, E, Nn, use_mask);

  // Scalar loss
  gnn_finalize<<<1, 64, 0, stream>>>(svec, scal, out, Nn);
}